// RNNCRFDecoder_51221779972194
// MI455X (gfx1250) — compile-verified
//
#include <hip/hip_runtime.h>
#include <hip/hip_bf16.h>

// Problem constants (from reference): B=128, T=1024, D=512, K(tags)=16
#define BB   128
#define TT   1024
#define DD   512
#define NT   16

typedef __attribute__((ext_vector_type(2))) float v2f;
typedef __attribute__((ext_vector_type(8))) float v8f;

// ---------------------------------------------------------------------------
// Kernel 1: logits[b,t,k] = sum_d H[b,t,d]*W[k,d] + bias[k]
// One wave computes one 16(T) x 16(K) tile, accumulating over D in K=4 chunks
// with V_WMMA_F32_16X16X4_F32 (full fp32 — matches reference precision).
//
// ISA 7.12.2 layouts (wave32):
//   A 16x4 f32 : lanes 0-15 hold M=0..15 {K=0,K=1} in v[0],v[1];
//                lanes 16-31 hold M=0..15 {K=2,K=3}.
//   B 4x16 f32 : lanes 0-15 hold N=0..15 rows {K=0,K=1}; lanes 16-31 {K=2,K=3}.
//   C/D 16x16  : VGPR r = row (r + 8*hi), col = lane&15.
// ---------------------------------------------------------------------------
__global__ __launch_bounds__(256) void logits_gemm_wmma(
    const float* __restrict__ H, const float* __restrict__ Wm,
    const float* __restrict__ bias, float* __restrict__ logits)
{
    const int wid  = (blockIdx.x * 256 + threadIdx.x) >> 5;  // global wave id
    const int lane = threadIdx.x & 31;
    const int b    = wid >> 6;          // batch
    const int t0   = (wid & 63) << 4;   // time-tile start
    const int row  = lane & 15;         // M index (A) == N index (B)
    const int hi   = lane >> 4;         // K sub-pair select

    const float* aptr = H  + ((long)b * TT + (t0 + row)) * DD + 2 * hi;
    const float* bptr = Wm + (long)row * DD + 2 * hi;   // B[k][n] = W[n][d0+k]

    v8f acc = {};
#pragma unroll 4
    for (int d0 = 0; d0 < DD; d0 += 4) {
        v2f a  = *(const v2f*)(aptr + d0);
        v2f bm = *(const v2f*)(bptr + d0);
        // (neg_a, A, neg_b, B, c_mod, C, reuse_a, reuse_b)
        acc = __builtin_amdgcn_wmma_f32_16x16x4_f32(
            false, a, false, bm, (short)0, acc, false, false);
    }

    const float bv = bias[row];
    const long  obase = ((long)b * TT + t0 + 8 * hi) * NT + row;
#pragma unroll
    for (int r = 0; r < 8; ++r)
        logits[obase + (long)r * NT] = acc[r] + bv;
}

// ---------------------------------------------------------------------------
// Kernel 2: CRF forward (logZ + Viterbi + numerator) + backtrace.
// One wave (32 lanes) per batch element:
//   lanes 0-15  : alpha recursion (logsumexp)        -> logZ
//   lanes 16-31 : delta recursion (max/argmax)       -> backpointers in LDS
// Cross-state broadcast via __shfl within each 16-lane half (ds_bpermute).
// Backtrace runs on lane 0 against LDS backpointers (short dependent chain).
// ---------------------------------------------------------------------------
__global__ __launch_bounds__(32) void crf_forward_bt(
    const float* __restrict__ logits, const int* __restrict__ mask,
    const int* __restrict__ labels, const float* __restrict__ start,
    const float* __restrict__ trans, const float* __restrict__ endv,
    float* __restrict__ pred, float* __restrict__ diff)
{
    __shared__ unsigned char hist[(TT - 1) * NT];   // ~16 KB backpointers

    const int b    = blockIdx.x;
    const int lane = threadIdx.x;
    const int j    = lane & 15;      // state handled by this lane
    const int half = lane & 16;      // 0 = logsumexp half, 16 = viterbi half

    const float* lg   = logits + (long)b * TT * NT;
    const int*   mrow = mask   + (long)b * TT;
    const int*   lrow = labels + (long)b * TT;

    float tr[NT];                    // column trans[:, j]
#pragma unroll
    for (int i = 0; i < NT; ++i) tr[i] = trans[i * NT + j];

    float score = start[j] + lg[j];

    for (int t = 1; t < TT; ++t) {
        const float e = lg[t * NT + j];
        const int   m = mrow[t];

        float xs[NT];
        float mx  = -3.402823466e38f;
        int   arg = 0;
#pragma unroll
        for (int i = 0; i < NT; ++i) {
            float v = __shfl(score, half | i, 32) + tr[i];
            xs[i] = v;
            if (v > mx) { mx = v; arg = i; }   // strict > => first-max (jnp.argmax)
        }

        float nxt;
        if (half == 0) {                       // logsumexp half
            float s = 0.f;
#pragma unroll
            for (int i = 0; i < NT; ++i) s += __expf(xs[i] - mx);
            nxt = mx + __logf(s) + e;
        } else {                               // viterbi half
            nxt = mx + e;
            hist[(t - 1) * NT + j] = (unsigned char)arg;  // recorded regardless of mask
        }
        if (m) score = nxt;
    }

    const float fin = score + endv[j];

    // half 0: logZ = logsumexp over 16 lanes (xor offsets stay within half)
    float m2 = fin;
    for (int off = 1; off < 16; off <<= 1) m2 = fmaxf(m2, __shfl_xor(m2, off, 32));
    float ss = __expf(fin - m2);
    for (int off = 1; off < 16; off <<= 1) ss += __shfl_xor(ss, off, 32);
    const float logZ = m2 + __logf(ss);        // valid in lanes 0-15 (lane 0 used)

    // half 1: argmax (first-max tie rule)
    float bv = fin; int bi = j;
    for (int off = 1; off < 16; off <<= 1) {
        float ov = __shfl_xor(bv, off, 32);
        int   oi = __shfl_xor(bi, off, 32);
        if (ov > bv || (ov == bv && oi < bi)) { bv = ov; bi = oi; }
    }
    const int best = __shfl(bi, 16, 32);

    // numerator: fully parallel masked sum over t, then wave reduce
    float nsum = 0.f; int msum = 0;
    for (int t = lane; t < TT; t += 32) {
        const int mt = mrow[t];
        msum += mt;
        if (t >= 1 && mt) {
            const int tp = lrow[t - 1], tc = lrow[t];
            nsum += trans[tp * NT + tc] + lg[t * NT + tc];
        }
    }
    for (int off = 1; off < 32; off <<= 1) {
        nsum += __shfl_xor(nsum, off, 32);
        msum += __shfl_xor(msum, off, 32);
    }

    __syncthreads();  // make hist[] LDS writes visible for backtrace

    if (lane == 0) {
        const int t0tag = lrow[0];
        float num = nsum + start[t0tag] + lg[t0tag];
        num += endv[lrow[msum - 1]];           // end[last_tag]
        diff[b] = num - logZ;

        // backtrace (lane 0, LDS-latency dependent chain)
        int cur = best;
        for (int t = TT - 1; t >= 1; --t) {
            const int m = mrow[t];
            pred[(long)b * TT + t] = (float)(m ? cur : 0);
            const int prev = hist[(t - 1) * NT + cur];
            if (m) cur = prev;
        }
        pred[(long)b * TT] = (float)cur;
    }
}

// ---------------------------------------------------------------------------
// Kernel 3: deterministic fixed-order loss reduction: loss = -mean(diff)
// ---------------------------------------------------------------------------
__global__ void finalize_loss(const float* __restrict__ diff, float* __restrict__ out_loss)
{
    if (blockIdx.x == 0 && threadIdx.x == 0) {
        float s = 0.f;
        for (int b = 0; b < BB; ++b) s += diff[b];
        out_loss[0] = -s / (float)BB;
    }
}

extern "C" void kernel_launch(void* const* d_in, const int* in_sizes, int n_in,
                              void* d_out, int out_size, void* d_ws, size_t ws_size,
                              hipStream_t stream)
{
    const float* H      = (const float*)d_in[0];   // (B,T,D) f32
    const int*   mask   = (const int*)  d_in[1];   // (B,T)   int (bool)
    const int*   labels = (const int*)  d_in[2];   // (B,T)   int
    const float* Wm     = (const float*)d_in[3];   // (K,D)   f32
    const float* bias   = (const float*)d_in[4];   // (K,)
    const float* start  = (const float*)d_in[5];   // (K,)
    const float* trans  = (const float*)d_in[6];   // (K,K)
    const float* endv   = (const float*)d_in[7];   // (K,)

    float* logits = (float*)d_ws;                                        // 8 MB
    float* diff   = (float*)((char*)d_ws + (size_t)BB * TT * NT * sizeof(float));

    float* pred = (float*)d_out;              // B*T tags (as float)
    float* loss = pred + (size_t)BB * TT;     // scalar loss

    // 8192 tiles, 8 waves (256 threads) per block -> 1024 blocks
    logits_gemm_wmma<<<(BB * (TT / 16)) / 8, 256, 0, stream>>>(H, Wm, bias, logits);
    crf_forward_bt<<<BB, 32, 0, stream>>>(logits, mask, labels, start, trans, endv,
                                          pred, diff);
    finalize_loss<<<1, 32, 0, stream>>>(diff, loss);
}